// SimpleAttentionLayer_86260123174625
// MI455X (gfx1250) — compile-verified
//
#include <hip/hip_runtime.h>
#include <hip/hip_bf16.h>

#define N_NODES 50000
#define N_EDGES 600000
#define D 128
#define LN_EPS 1e-5f
#define SLOPE 0.2f

typedef __attribute__((ext_vector_type(2))) float v2f;
typedef __attribute__((ext_vector_type(4))) float v4f;
typedef __attribute__((ext_vector_type(8))) float v8f;

// ---------- monotonic float<->uint map for atomic max on possibly-negative floats ----------
__device__ __forceinline__ unsigned fmax_map(float f) {
    unsigned u = __float_as_uint(f);
    return (u & 0x80000000u) ? ~u : (u | 0x80000000u);
}
__device__ __forceinline__ float fmax_unmap(unsigned u) {
    u = (u & 0x80000000u) ? (u & 0x7FFFFFFFu) : ~u;
    return __uint_as_float(u);
}

// ---------- K0: zero the output accumulator + per-node softmax state ----------
__global__ __launch_bounds__(256) void k_init(float* __restrict__ out,
                                              float* __restrict__ z,
                                              unsigned* __restrict__ m) {
    int i = blockIdx.x * blockDim.x + threadIdx.x;
    if (i < N_NODES * D) out[i] = 0.0f;
    if (i < N_NODES) { z[i] = 0.0f; m[i] = 0u; }   // mapped 0 < mapped(-inf)
}

// ---------- K1: h = x @ W^T via V_WMMA_F32_16X16X4_F32, one wave per 16x16 tile ----------
__global__ __launch_bounds__(256) void k_gemm_wmma(const float* __restrict__ x,
                                                   const float* __restrict__ W,
                                                   float* __restrict__ h) {
    int wave = (blockIdx.x * blockDim.x + threadIdx.x) >> 5;
    int lane = threadIdx.x & 31;
    const int tiles_n = D / 16;                 // 8
    int tm = wave / tiles_n;                    // 0..3124
    int tn = wave - tm * tiles_n;
    if (tm >= N_NODES / 16) return;             // wave-uniform; grid is exact anyway
    int m0 = tm * 16, n0 = tn * 16;

    int l15  = lane & 15;
    int koff = (lane >> 4) * 2;                 // lane half selects K pair {0,1} vs {2,3}
    const float* arow = x + (size_t)(m0 + l15) * D + koff;   // A: M on lanes, K in VGPRs
    const float* brow = W + (size_t)(n0 + l15) * D + koff;   // B[k][n] = W[n][k]

    v8f c = {};
#pragma unroll
    for (int k = 0; k < D; k += 4) {
        v2f a = *(const v2f*)(arow + k);
        v2f b = *(const v2f*)(brow + k);
        // (neg_a, A, neg_b, B, c_mod, C, reuse_a, reuse_b)
        c = __builtin_amdgcn_wmma_f32_16x16x4_f32(false, a, false, b,
                                                  (short)0, c, false, false);
    }

    int mbase = m0 + (lane >> 4) * 8;           // C/D: VGPR r -> rows r and r+8
#pragma unroll
    for (int r = 0; r < 8; ++r)
        h[(size_t)(mbase + r) * D + n0 + l15] = c[r];
}

// ---------- K2: s_src[n] = h[n]·a[:128], s_dst[n] = h[n]·a[128:], wave per node ----------
__global__ __launch_bounds__(256) void k_scores(const float* __restrict__ h,
                                                const float* __restrict__ a,
                                                float* __restrict__ s_src,
                                                float* __restrict__ s_dst) {
    int node = (blockIdx.x * blockDim.x + threadIdx.x) >> 5;
    int lane = threadIdx.x & 31;
    if (node >= N_NODES) return;
    v4f hv = *(const v4f*)(h + (size_t)node * D + lane * 4);
    v4f a0 = *(const v4f*)(a + lane * 4);
    v4f a1 = *(const v4f*)(a + D + lane * 4);
    float ss = hv.x * a0.x + hv.y * a0.y + hv.z * a0.z + hv.w * a0.w;
    float sd = hv.x * a1.x + hv.y * a1.y + hv.z * a1.z + hv.w * a1.w;
#pragma unroll
    for (int off = 16; off > 0; off >>= 1) {
        ss += __shfl_xor(ss, off, 32);
        sd += __shfl_xor(sd, off, 32);
    }
    if (lane == 0) { s_src[node] = ss; s_dst[node] = sd; }
}

// ---------- K3: per-edge leaky-relu score, segment max via mapped atomicMax ----------
__global__ __launch_bounds__(256) void k_edge_max(const long long* __restrict__ src,
                                                  const long long* __restrict__ dst,
                                                  const float* __restrict__ s_src,
                                                  const float* __restrict__ s_dst,
                                                  unsigned* __restrict__ m) {
    int i = blockIdx.x * blockDim.x + threadIdx.x;
    if (i >= N_EDGES) return;
    int s = (int)src[i], d = (int)dst[i];
    float e = s_src[s] + s_dst[d];
    e = (e > 0.0f) ? e : SLOPE * e;
    atomicMax(m + d, fmax_map(e));
}

// ---------- K4: ex = exp(e - m[dst]); z[dst] += ex ----------
__global__ __launch_bounds__(256) void k_edge_exp(const long long* __restrict__ src,
                                                  const long long* __restrict__ dst,
                                                  const float* __restrict__ s_src,
                                                  const float* __restrict__ s_dst,
                                                  const unsigned* __restrict__ m,
                                                  float* __restrict__ z,
                                                  float* __restrict__ ex) {
    int i = blockIdx.x * blockDim.x + threadIdx.x;
    if (i >= N_EDGES) return;
    int s = (int)src[i], d = (int)dst[i];
    float e = s_src[s] + s_dst[d];
    e = (e > 0.0f) ? e : SLOPE * e;
    float v = __expf(e - fmax_unmap(m[d]));
    ex[i] = v;
    atomicAdd(z + d, v);
}

// ---------- K5: out[dst] += alpha * h[src], wave per edge, 4 dims per lane ----------
__global__ __launch_bounds__(256) void k_edge_scatter(const long long* __restrict__ src,
                                                      const long long* __restrict__ dst,
                                                      const float* __restrict__ ex,
                                                      const float* __restrict__ z,
                                                      const float* __restrict__ h,
                                                      float* __restrict__ out) {
    int edge = (blockIdx.x * blockDim.x + threadIdx.x) >> 5;
    int lane = threadIdx.x & 31;
    if (edge >= N_EDGES) return;
    int s = (int)src[edge], d = (int)dst[edge];
    float alpha = ex[edge] / z[d];
    v4f hv = *(const v4f*)(h + (size_t)s * D + lane * 4);
    float* o = out + (size_t)d * D + lane * 4;
    atomicAdd(o + 0, hv.x * alpha);
    atomicAdd(o + 1, hv.y * alpha);
    atomicAdd(o + 2, hv.z * alpha);
    atomicAdd(o + 3, hv.w * alpha);
}

// ---------- K6: y = agg + x; LayerNorm; wave per node ----------
__global__ __launch_bounds__(256) void k_finalize(const float* __restrict__ x,
                                                  const float* __restrict__ gamma,
                                                  const float* __restrict__ beta,
                                                  float* __restrict__ out) {
    int node = (blockIdx.x * blockDim.x + threadIdx.x) >> 5;
    int lane = threadIdx.x & 31;
    if (node >= N_NODES) return;
    float* orow = out + (size_t)node * D + lane * 4;
    v4f agg = *(const v4f*)orow;
    v4f xv  = *(const v4f*)(x + (size_t)node * D + lane * 4);
    v4f y   = agg + xv;
    float sum = y.x + y.y + y.z + y.w;
    float sq  = y.x * y.x + y.y * y.y + y.z * y.z + y.w * y.w;
#pragma unroll
    for (int off = 16; off > 0; off >>= 1) {
        sum += __shfl_xor(sum, off, 32);
        sq  += __shfl_xor(sq,  off, 32);
    }
    float mu  = sum * (1.0f / D);
    float var = sq * (1.0f / D) - mu * mu;
    float inv = rsqrtf(var + LN_EPS);
    v4f g = *(const v4f*)(gamma + lane * 4);
    v4f b = *(const v4f*)(beta + lane * 4);
    orow[0] = (y.x - mu) * inv * g.x + b.x;
    orow[1] = (y.y - mu) * inv * g.y + b.y;
    orow[2] = (y.z - mu) * inv * g.z + b.z;
    orow[3] = (y.w - mu) * inv * g.w + b.w;
}

extern "C" void kernel_launch(void* const* d_in, const int* in_sizes, int n_in,
                              void* d_out, int out_size, void* d_ws, size_t ws_size,
                              hipStream_t stream) {
    (void)in_sizes; (void)n_in; (void)out_size; (void)ws_size;

    const float*     x     = (const float*)d_in[0];
    const long long* eidx  = (const long long*)d_in[1];   // int64 [2, E]
    const float*     W     = (const float*)d_in[2];
    const float*     avec  = (const float*)d_in[3];
    const float*     gamma = (const float*)d_in[4];
    const float*     beta  = (const float*)d_in[5];
    float*           out   = (float*)d_out;

    const long long* src = eidx;
    const long long* dst = eidx + N_EDGES;

    // workspace layout (floats): h | s_src | s_dst | m | z | ex
    float* ws    = (float*)d_ws;
    float* h     = ws;                        // 6,400,000
    float* s_src = h + (size_t)N_NODES * D;   // 50,000
    float* s_dst = s_src + N_NODES;           // 50,000
    unsigned* m  = (unsigned*)(s_dst + N_NODES); // 50,000
    float* z     = (float*)(m + N_NODES);     // 50,000
    float* ex    = z + N_NODES;               // 600,000

    // K0: init accumulators
    k_init<<<(N_NODES * D + 255) / 256, 256, 0, stream>>>(out, z, m);

    // K1: GEMM via f32 WMMA — 25000 waves (3125x8 tiles), 8 waves/block
    k_gemm_wmma<<<(N_NODES / 16) * (D / 16) / 8, 256, 0, stream>>>(x, W, h);

    // K2: per-node attention scores — 50000 waves
    k_scores<<<N_NODES / 8, 256, 0, stream>>>(h, avec, s_src, s_dst);

    // K3/K4: segment softmax over dst
    k_edge_max<<<(N_EDGES + 255) / 256, 256, 0, stream>>>(src, dst, s_src, s_dst, m);
    k_edge_exp<<<(N_EDGES + 255) / 256, 256, 0, stream>>>(src, dst, s_src, s_dst, m, z, ex);

    // K5: weighted scatter-add — wave per edge
    k_edge_scatter<<<N_EDGES / 8, 256, 0, stream>>>(src, dst, ex, z, h, out);

    // K6: residual + LayerNorm — wave per node
    k_finalize<<<N_NODES / 8, 256, 0, stream>>>(x, gamma, beta, out);
}